// G2O_56882546868900
// MI455X (gfx1250) — compile-verified
//
#include <hip/hip_runtime.h>
#include <hip/hip_bf16.h>
#include <math.h>

// Problem constants (match reference)
#define NG    4
#define NBOX  64
#define FG    1024   // NFG
#define FR    256    // NFR
#define BATCH 2048
#define SCALE_F 0.04419417382415922f   // 1/sqrt(256+256)
#define LN_EPS  1e-5f

typedef float v2f __attribute__((ext_vector_type(2)));
typedef float v8f __attribute__((ext_vector_type(8)));

// ---------------------------------------------------------------------------
// CDNA5 async global->LDS helpers (ASYNCcnt path, bypasses VGPRs).
// Generic pointers to LDS carry the DS byte offset in their low 32 bits.
// ---------------------------------------------------------------------------
__device__ __forceinline__ uint32_t lds_addr_of(const void* p)
{
    return (uint32_t)(uintptr_t)p;
}
__device__ __forceinline__ void async_b128(uint32_t lds_dst, const float* gsrc)
{
    asm volatile("global_load_async_to_lds_b128 %0, %1, off"
                 :: "v"(lds_dst), "v"(gsrc) : "memory");
}
__device__ __forceinline__ void wait_async0()
{
    asm volatile("s_wait_asynccnt 0" ::: "memory");
}

// ---------------------------------------------------------------------------
// Kernel 1: xs[b,f] = sum_n x[b,n,f]   (streams x once, 512 MiB)
// ---------------------------------------------------------------------------
__global__ __launch_bounds__(256)
void sum_boxes_kernel(const float* __restrict__ x, float* __restrict__ xs)
{
    const int b = blockIdx.x;
    const int t = threadIdx.x;                // f4 = 4*t, 256 threads cover 1024 f
    const float4* xp = (const float4*)(x + (long)b * NBOX * FG) + t;
    float4 acc = make_float4(0.f, 0.f, 0.f, 0.f);
    #pragma unroll 4
    for (int n = 0; n < NBOX; ++n) {
        float4 vv = xp[n * (FG / 4)];
        acc.x += vv.x; acc.y += vv.y; acc.z += vv.z; acc.w += vv.w;
    }
    ((float4*)(xs + (long)b * FG))[t] = acc;
}

// ---------------------------------------------------------------------------
// Batched WMMA f32 GEMM:  C[g] (M x N) = A[g] (M x K) * B[g] (K x N) (+bias)
//   WLAYOUT_NK == 1 : W stored row-major [N,K]  (nn.Linear weight, W[n*K+k])
//   WLAYOUT_NK == 0 : W stored row-major [K,N]  (W[k*N+n])
// Block tile 64x128, K-chunk 32, 8 waves; each wave -> 16x64 output via
// 4 accumulators of v_wmma_f32_16x16x4_f32.
// Tiles are staged with double-buffered global_load_async_to_lds_b128 so the
// next chunk's fill overlaps the current chunk's WMMA work.
// ---------------------------------------------------------------------------
template<int WLAYOUT_NK>
__global__ __launch_bounds__(256)
void gemm_wmma_f32(const float* __restrict__ A, const float* __restrict__ W,
                   const float* __restrict__ bias, float bias_scale,
                   float* __restrict__ C,
                   int M, int Ntot, int K, long a_gstride)
{
    constexpr int MB = 64, NB = 128, KC = 32;
    constexpr int LDA    = KC + 4;   // 36 floats: 144B rows -> 16B-aligned B128 dst
    constexpr int LDB_NK = KC + 4;   // 36 floats, n-major [n][k]
    constexpr int LDB_KN = NB + 8;   // 136 floats, k-major [k][n], conflict-free halves
    __shared__ float As[2][MB * LDA];
    __shared__ float Bs[2][(WLAYOUT_NK ? NB * LDB_NK : KC * LDB_KN)];

    const int g = blockIdx.z;
    A += (long)g * a_gstride;
    W += (long)g * (long)Ntot * (long)K;
    C += (long)g * (long)M * (long)Ntot;

    const int m0 = blockIdx.x * MB;
    const int n0 = blockIdx.y * NB;
    const int t    = threadIdx.x;
    const int lane = t & 31;
    const int wave = t >> 5;
    const int wm   = wave & 3;        // wave row   (4 x 16 = 64 rows)
    const int wn   = wave >> 2;       // wave col   (2 x 64 = 128 cols)
    const int lhi  = lane >> 4;       // K-half select (ISA A/B f32 layout)
    const int llo  = lane & 15;       // M (A) / N (B) within fragment

    // per-thread staging coordinates
    const int ar = t >> 2;            // A: 0..63
    const int ac = (t & 3) * 8;       // A: 0,8,16,24  (2 x B128)

    auto stage = [&](int buf, int k0) {
        {
            const float* ap = A + (long)(m0 + ar) * K + k0 + ac;
            uint32_t al = lds_addr_of(&As[buf][ar * LDA + ac]);
            async_b128(al,      ap);
            async_b128(al + 16, ap + 4);
        }
        if constexpr (WLAYOUT_NK) {
            const int n  = t >> 1;                 // 0..127
            const int cb = (t & 1) * 16;           // 4 x B128
            const float* wp = W + (long)(n0 + n) * K + k0 + cb;
            uint32_t bl = lds_addr_of(&Bs[buf][n * LDB_NK + cb]);
            #pragma unroll
            for (int i = 0; i < 4; ++i) async_b128(bl + 16 * i, wp + 4 * i);
        } else {
            const int k  = t >> 3;                 // 0..31
            const int nb = (t & 7) * 16;           // 4 x B128
            const float* wp = W + (long)(k0 + k) * Ntot + n0 + nb;
            uint32_t bl = lds_addr_of(&Bs[buf][k * LDB_KN + nb]);
            #pragma unroll
            for (int i = 0; i < 4; ++i) async_b128(bl + 16 * i, wp + 4 * i);
        }
    };

    v8f acc[4];
    #pragma unroll
    for (int j = 0; j < 4; ++j) acc[j] = (v8f)0.0f;

    stage(0, 0);                       // prologue fill
    int buf = 0;
    for (int k0 = 0; k0 < K; k0 += KC) {
        wait_async0();                 // my async fills for `buf` landed
        __syncthreads();               // everyone's fills landed; prev reads done
        if (k0 + KC < K) stage(buf ^ 1, k0 + KC);   // prefetch next chunk

        #pragma unroll
        for (int kk = 0; kk < KC; kk += 4) {
            // A 16x4 f32 fragment: lanes 0-15 -> K=0,1 ; lanes 16-31 -> K=2,3
            v2f a = *(const v2f*)&As[buf][(wm * 16 + llo) * LDA + kk + lhi * 2];
            #pragma unroll
            for (int j = 0; j < 4; ++j) {
                v2f bf;
                if constexpr (WLAYOUT_NK) {
                    bf = *(const v2f*)&Bs[buf][(wn * 64 + j * 16 + llo) * LDB_NK + kk + lhi * 2];
                } else {
                    const int nn = wn * 64 + j * 16 + llo;
                    bf[0] = Bs[buf][(kk + lhi * 2    ) * LDB_KN + nn];
                    bf[1] = Bs[buf][(kk + lhi * 2 + 1) * LDB_KN + nn];
                }
                acc[j] = __builtin_amdgcn_wmma_f32_16x16x4_f32(
                    /*neg_a=*/false, a, /*neg_b=*/false, bf,
                    /*c_mod=*/(short)0, acc[j], /*reuse_a=*/false, /*reuse_b=*/false);
            }
        }
        buf ^= 1;
    }

    // --- epilogue: C/D layout: m = vgpr + 8*(lane>>4), n = lane&15 ---
    #pragma unroll
    for (int j = 0; j < 4; ++j) {
        const int n = wn * 64 + j * 16 + llo;
        float bv = 0.f;
        if (bias) bv = bias_scale * bias[(long)g * Ntot + n0 + n];
        #pragma unroll
        for (int vv = 0; vv < 8; ++vv) {
            const int m = wm * 16 + vv + 8 * lhi;
            C[(long)(m0 + m) * Ntot + n0 + n] = acc[j][vv] + bv;
        }
    }
}

// ---------------------------------------------------------------------------
// Kernel 4: fused relation + softmax + aggregation, one block per batch b.
//   rel[g,n] = SCALE * x[b,n,:]·v[g,b,:]   (bt bias cancels in softmax)
//   w = softmax_n(rel);  agg[g,b,f] = sum_n w[g,n] * x[b,n,f]
// x[b] (256 KiB) is streamed in 64-wide f-chunks through LDS; the second
// pass re-reads it (hits L2: same block touches it back-to-back).
// ---------------------------------------------------------------------------
__global__ __launch_bounds__(256)
void attn_fused_kernel(const float* __restrict__ x, const float* __restrict__ v,
                       float* __restrict__ agg)
{
    constexpr int FC = 64, LDX = FC + 1;
    __shared__ float xsh[NBOX * LDX];   // [n][f_local]
    __shared__ float vsh[NG * FC];      // [g][f_local]
    __shared__ float wsh[NG * NBOX];    // softmax weights
    __shared__ float tsh[NG * NBOX];    // raw relation

    const int b = blockIdx.x;
    const int t = threadIdx.x;
    const int g = t >> 6;               // 0..3
    const int q = t & 63;               // pass1: n ; pass2: f_local

    // ---- pass 1: relation ----
    float racc = 0.0f;
    for (int f0 = 0; f0 < FG; f0 += FC) {
        {
            const int n  = t >> 2;
            const int fb = (t & 3) * 16;
            const float* src = x + ((long)b * NBOX + n) * FG + f0 + fb;
            #pragma unroll
            for (int i = 0; i < 16; ++i) xsh[n * LDX + fb + i] = src[i];
        }
        vsh[t] = v[((long)g * BATCH + b) * FG + f0 + q];
        __syncthreads();
        float p = 0.f;
        #pragma unroll 8
        for (int f = 0; f < FC; ++f) p += xsh[q * LDX + f] * vsh[g * FC + f];
        racc += p;
        __syncthreads();
    }
    racc *= SCALE_F;

    // ---- softmax over n (64) within each g ----
    tsh[t] = racc;
    __syncthreads();
    float mx = -3.0e38f;
    for (int i = 0; i < NBOX; ++i) mx = fmaxf(mx, tsh[g * NBOX + i]);
    float e = expf(racc - mx);
    wsh[t] = e;
    __syncthreads();
    float s = 0.f;
    for (int i = 0; i < NBOX; ++i) s += wsh[g * NBOX + i];
    __syncthreads();                    // all reads of wsh done before overwrite
    wsh[t] = e / s;
    __syncthreads();

    // ---- pass 2: aggregation ----
    for (int f0 = 0; f0 < FG; f0 += FC) {
        {
            const int n  = t >> 2;
            const int fb = (t & 3) * 16;
            const float* src = x + ((long)b * NBOX + n) * FG + f0 + fb;
            #pragma unroll
            for (int i = 0; i < 16; ++i) xsh[n * LDX + fb + i] = src[i];
        }
        __syncthreads();
        float a = 0.f;
        #pragma unroll 8
        for (int n = 0; n < NBOX; ++n) a += wsh[g * NBOX + n] * xsh[n * LDX + q];
        agg[((long)g * BATCH + b) * FG + f0 + q] = a;
        __syncthreads();
    }
}

// ---------------------------------------------------------------------------
// Kernel 6: LayerNorm (per g) + affine + ReLU + mean over the NG graphs.
// One block per b; each thread owns 4 features (t, t+256, t+512, t+768).
// ---------------------------------------------------------------------------
__global__ __launch_bounds__(256)
void ln_mean_kernel(const float* __restrict__ h, const float* __restrict__ gamma,
                    const float* __restrict__ beta, float* __restrict__ out)
{
    __shared__ float red [256];
    __shared__ float red2[256];
    const int b = blockIdx.x;
    const int t = threadIdx.x;
    float o[4] = {0.f, 0.f, 0.f, 0.f};

    for (int g = 0; g < NG; ++g) {
        float hv[4], s = 0.f, s2 = 0.f;
        #pragma unroll
        for (int i = 0; i < 4; ++i) {
            hv[i] = h[((long)g * BATCH + b) * FG + t + 256 * i];
            s  += hv[i];
            s2 += hv[i] * hv[i];
        }
        red[t] = s; red2[t] = s2;
        __syncthreads();
        for (int st = 128; st > 0; st >>= 1) {
            if (t < st) { red[t] += red[t + st]; red2[t] += red2[t + st]; }
            __syncthreads();
        }
        const float mu   = red[0]  * (1.0f / FG);
        const float var  = red2[0] * (1.0f / FG) - mu * mu;
        const float rinv = rsqrtf(var + LN_EPS);
        __syncthreads();                 // red[] reused next g
        #pragma unroll
        for (int i = 0; i < 4; ++i) {
            const int f = t + 256 * i;
            float nv = (hv[i] - mu) * rinv * gamma[g * FG + f] + beta[g * FG + f];
            o[i] += fmaxf(nv, 0.f) * (1.0f / NG);
        }
    }
    #pragma unroll
    for (int i = 0; i < 4; ++i) out[(long)b * FG + t + 256 * i] = o[i];
}

// ---------------------------------------------------------------------------
// Launch
// ---------------------------------------------------------------------------
extern "C" void kernel_launch(void* const* d_in, const int* in_sizes, int n_in,
                              void* d_out, int out_size, void* d_ws, size_t ws_size,
                              hipStream_t stream)
{
    (void)in_sizes; (void)n_in; (void)out_size; (void)ws_size;
    const float* x     = (const float*)d_in[0];
    const float* Wt    = (const float*)d_in[1];
    /* d_in[2] = bt: a uniform shift over n inside softmax -> drops out */
    const float* Wp    = (const float*)d_in[3];
    const float* bp    = (const float*)d_in[4];
    const float* Wg    = (const float*)d_in[5];
    const float* gamma = (const float*)d_in[6];
    const float* beta  = (const float*)d_in[7];
    float* out = (float*)d_out;

    float* ws     = (float*)d_ws;
    float* xs     = ws;                                  // [B,FG]        8 MiB
    float* phisum = xs     + (size_t)BATCH * FG;         // [G,B,FR]      8 MiB
    float* vbuf   = phisum + (size_t)NG * BATCH * FR;    // [G,B,FG]     32 MiB
    float* aggbuf = vbuf   + (size_t)NG * BATCH * FG;    // [G,B,FG]     32 MiB
    float* hbuf   = aggbuf + (size_t)NG * BATCH * FG;    // [G,B,FG]     32 MiB

    // 1) xs = sum over boxes
    sum_boxes_kernel<<<BATCH, 256, 0, stream>>>(x, xs);

    // 2) phi_sum[g] = xs * Wp[g]^T + NBOX*bp[g]   (Wp: [FR,FG] row-major = [N,K])
    gemm_wmma_f32<1><<<dim3(BATCH / 64, FR / 128, NG), 256, 0, stream>>>(
        xs, Wp, bp, (float)NBOX, phisum, BATCH, FR, FG, /*a_gstride=*/0L);

    // 3) v[g] = phi_sum[g] * Wt[g]                (Wt: [FR,FG] row-major = [K,N])
    gemm_wmma_f32<0><<<dim3(BATCH / 64, FG / 128, NG), 256, 0, stream>>>(
        phisum, Wt, nullptr, 0.f, vbuf, BATCH, FG, FR, (long)BATCH * FR);

    // 4) relation -> softmax -> aggregation (streams x, 2nd pass L2-resident)
    attn_fused_kernel<<<BATCH, 256, 0, stream>>>(x, vbuf, aggbuf);

    // 5) h[g] = agg[g] * Wg[g]^T                  (Wg: [FG,FG] row-major = [N,K])
    gemm_wmma_f32<1><<<dim3(BATCH / 64, FG / 128, NG), 256, 0, stream>>>(
        aggbuf, Wg, nullptr, 0.f, hbuf, BATCH, FG, FG, (long)BATCH * FG);

    // 6) LayerNorm + ReLU + mean over graphs -> out [B,1,FG]
    ln_mean_kernel<<<BATCH, 256, 0, stream>>>(hbuf, gamma, beta, out);
}